// DirectedEdgeConv_44650480009512
// MI455X (gfx1250) — compile-verified
//
#include <hip/hip_runtime.h>

// Problem constants pinned by the reference.
#define NN 50000      // num_nodes
#define DD 128        // feature dim
#define TILE_E 32     // edges per block tile (2 WMMA M-tiles)
#define PITCH 132     // LDS row pitch in floats (128 + 4) -> conflict-free ds_load_b64

typedef float v2f __attribute__((ext_vector_type(2)));
typedef float v8f __attribute__((ext_vector_type(8)));

// ---------------- Kernel 1: zero the workspace (sums + counts) ----------------
__global__ void zero_ws_kernel(float* __restrict__ ws, int n) {
    int i = blockIdx.x * blockDim.x + threadIdx.x;
    if (i < n) ws[i] = 0.0f;
}

// ---------------- Kernel 2: scatter-add edge rows into node sums --------------
// 32 threads per edge, 4 floats per thread; sums live in L2 (51 MB << 192 MB).
__global__ void scatter_accum_kernel(const float* __restrict__ ea,
                                     const long long* __restrict__ src,
                                     const long long* __restrict__ dst,
                                     float* __restrict__ sum_in,
                                     float* __restrict__ sum_out,
                                     float* __restrict__ cnt_in,
                                     float* __restrict__ cnt_out,
                                     int E) {
    int t = blockIdx.x * blockDim.x + threadIdx.x;
    int e = t >> 5;
    if (e >= E) return;
    int c = (t & 31) << 2;
    int s = (int)src[e];
    int d = (int)dst[e];
    float4 v = *(const float4*)(ea + (size_t)e * DD + c);
    float* pi = sum_in  + (size_t)d * DD + c;
    float* po = sum_out + (size_t)s * DD + c;
    atomicAdd(pi + 0, v.x); atomicAdd(pi + 1, v.y);
    atomicAdd(pi + 2, v.z); atomicAdd(pi + 3, v.w);
    atomicAdd(po + 0, v.x); atomicAdd(po + 1, v.y);
    atomicAdd(po + 2, v.z); atomicAdd(po + 3, v.w);
    if ((t & 31) == 0) {
        atomicAdd(cnt_in + d, 1.0f);
        atomicAdd(cnt_out + s, 1.0f);
    }
}

// ---------------- Kernel 3: sums -> means (divide by max(cnt,1)) --------------
__global__ void normalize_kernel(float* __restrict__ sums,
                                 const float* __restrict__ cnts,
                                 int total_quads) {
    int t = blockIdx.x * blockDim.x + threadIdx.x;
    if (t >= total_quads) return;
    int node = t >> 5;            // combined node index in [0, 2N)
    int c = (t & 31) << 2;
    float inv = 1.0f / fmaxf(cnts[node], 1.0f);
    float4* p = (float4*)(sums + (size_t)node * DD + c);
    float4 v = *p;
    v.x *= inv; v.y *= inv; v.z *= inv; v.w *= inv;
    *p = v;
}

// ---------------- Kernel 4: fused triple-GEMM with V_WMMA_F32_16X16X4_F32 ----
// Weights LDS-resident (198 KB); 32 edges/tile; each wave owns one 16-col
// n-tile and two independent accumulators (two M-tiles sharing B fragments).
__launch_bounds__(256, 1)
__global__ void fused_edge_gemm_kernel(const float* __restrict__ ea,
                                       const long long* __restrict__ src,
                                       const long long* __restrict__ dst,
                                       const float* __restrict__ mean_in,
                                       const float* __restrict__ mean_out,
                                       const float* __restrict__ Wself,
                                       const float* __restrict__ bias,
                                       const float* __restrict__ Win,
                                       const float* __restrict__ Wout,
                                       float* __restrict__ out,
                                       int E, int num_tiles) {
    __shared__ float sW[3][DD * PITCH];       // 3 x 128 x 132 f32 ~ 198 KB
    __shared__ float sA[3][TILE_E * PITCH];   // 3 x  32 x 132 f32 ~  50 KB
    __shared__ float sBias[DD];

    const int tid = threadIdx.x;

    // ---- Stage all three weight matrices into LDS once per block ----
    for (int f = tid; f < 3 * DD * 32; f += 256) {
        int m   = f / (DD * 32);
        int rem = f % (DD * 32);
        int row = rem >> 5;
        int c4  = (rem & 31) << 2;
        const float* wp = (m == 0) ? Wself : (m == 1) ? Win : Wout;
        float4 v = *(const float4*)(wp + row * DD + c4);
        *(float4*)(&sW[m][row * PITCH + c4]) = v;
    }
    if (tid < DD) sBias[tid] = bias[tid];
    __syncthreads();

    const int wave = tid >> 5;          // 8 waves: one 16-col n-tile each
    const int lane = tid & 31;
    const int half = lane >> 4;         // K-pair selector per WMMA layout
    const int mrow = lane & 15;         // A row / B,C column within tile
    const int ncol = wave * 16 + mrow;  // global output column

    // A fragment bases: matrix m, M-tile 0 (rows 0-15) and 1 (rows 16-31)
    const float* a00 = &sA[0][mrow * PITCH + 2 * half];
    const float* a01 = &sA[0][(16 + mrow) * PITCH + 2 * half];
    const float* a10 = &sA[1][mrow * PITCH + 2 * half];
    const float* a11 = &sA[1][(16 + mrow) * PITCH + 2 * half];
    const float* a20 = &sA[2][mrow * PITCH + 2 * half];
    const float* a21 = &sA[2][(16 + mrow) * PITCH + 2 * half];
    const float* bB0 = &sW[0][ncol * PITCH + 2 * half];
    const float* bB1 = &sW[1][ncol * PITCH + 2 * half];
    const float* bB2 = &sW[2][ncol * PITCH + 2 * half];

    for (int t = blockIdx.x; t < num_tiles; t += gridDim.x) {
        const long long e0 = (long long)t * TILE_E;

        // ---- Gather the three 32x128 A-tiles into LDS ----
        for (int f = tid; f < 3 * TILE_E * 32; f += 256) {
            int m   = f / (TILE_E * 32);
            int rem = f % (TILE_E * 32);
            int row = rem >> 5;
            int c4  = (rem & 31) << 2;
            long long e = e0 + row;
            if (e >= E) e = E - 1;
            const float* p;
            if (m == 0)      p = ea       + e * (long long)DD;
            else if (m == 1) p = mean_in  + (long long)src[e] * DD;
            else             p = mean_out + (long long)dst[e] * DD;
            *(float4*)(&sA[m][row * PITCH + c4]) = *(const float4*)(p + c4);
            // Prefetch next tile's sequential traffic (global_prefetch_b8)
            if (m == 0 && c4 == 0) {
                long long tn = (long long)t + gridDim.x;
                if (tn < num_tiles) {
                    long long en = tn * TILE_E + row;
                    if (en < E) {
                        __builtin_prefetch(ea + en * DD, 0, 1);
                        __builtin_prefetch(src + en, 0, 1);
                        __builtin_prefetch(dst + en, 0, 1);
                    }
                }
            }
        }
        __syncthreads();

        v8f acc0, acc1;                 // two independent WMMA chains
        float bv = sBias[ncol];
        #pragma unroll
        for (int i = 0; i < 8; ++i) { acc0[i] = bv; acc1[i] = bv; }

        #pragma unroll 2
        for (int k = 0; k < DD; k += 4) {
            v2f b0 = *(const v2f*)(bB0 + k);
            v2f x00 = *(const v2f*)(a00 + k);
            v2f x01 = *(const v2f*)(a01 + k);
            acc0 = __builtin_amdgcn_wmma_f32_16x16x4_f32(false, x00, false, b0,
                                                         (short)0, acc0, false, false);
            acc1 = __builtin_amdgcn_wmma_f32_16x16x4_f32(false, x01, false, b0,
                                                         (short)0, acc1, false, false);
            v2f b1 = *(const v2f*)(bB1 + k);
            v2f x10 = *(const v2f*)(a10 + k);
            v2f x11 = *(const v2f*)(a11 + k);
            acc0 = __builtin_amdgcn_wmma_f32_16x16x4_f32(false, x10, false, b1,
                                                         (short)0, acc0, false, false);
            acc1 = __builtin_amdgcn_wmma_f32_16x16x4_f32(false, x11, false, b1,
                                                         (short)0, acc1, false, false);
            v2f b2 = *(const v2f*)(bB2 + k);
            v2f x20 = *(const v2f*)(a20 + k);
            v2f x21 = *(const v2f*)(a21 + k);
            acc0 = __builtin_amdgcn_wmma_f32_16x16x4_f32(false, x20, false, b2,
                                                         (short)0, acc0, false, false);
            acc1 = __builtin_amdgcn_wmma_f32_16x16x4_f32(false, x21, false, b2,
                                                         (short)0, acc1, false, false);
        }
        __syncthreads();   // protect sA before next tile's staging

        // ---- leaky_relu(., 0.2) and store both C tiles ----
        // C layout: VGPR i -> row (i + 8*half), column ncol. 32-bit indexing.
        const int base = (int)e0 * DD + ncol;
        if (e0 + TILE_E <= (long long)E) {
            #pragma unroll
            for (int i = 0; i < 8; ++i) {
                float v0 = acc0[i];
                v0 = (v0 > 0.0f) ? v0 : 0.2f * v0;
                out[base + (i + 8 * half) * DD] = v0;
                float v1 = acc1[i];
                v1 = (v1 > 0.0f) ? v1 : 0.2f * v1;
                out[base + (16 + i + 8 * half) * DD] = v1;
            }
        } else {
            #pragma unroll
            for (int i = 0; i < 8; ++i) {
                int r0 = i + 8 * half;
                if (e0 + r0 < E) {
                    float v0 = acc0[i];
                    out[base + r0 * DD] = (v0 > 0.0f) ? v0 : 0.2f * v0;
                }
                int r1 = 16 + i + 8 * half;
                if (e0 + r1 < E) {
                    float v1 = acc1[i];
                    out[base + r1 * DD] = (v1 > 0.0f) ? v1 : 0.2f * v1;
                }
            }
        }
    }
}

extern "C" void kernel_launch(void* const* d_in, const int* in_sizes, int n_in,
                              void* d_out, int out_size, void* d_ws, size_t ws_size,
                              hipStream_t stream) {
    (void)n_in; (void)out_size; (void)ws_size;
    const float*      edge_attr = (const float*)d_in[0];
    const long long*  edge_idx  = (const long long*)d_in[1];   // int64 [2,E]
    // d_in[2] = num_nodes scalar (problem pins N = 50000)
    const float*      W_self_w  = (const float*)d_in[3];
    const float*      W_self_b  = (const float*)d_in[4];
    const float*      W_in_w    = (const float*)d_in[5];
    const float*      W_out_w   = (const float*)d_in[6];
    float*            out       = (float*)d_out;

    const int E = in_sizes[0] / DD;
    const long long* src = edge_idx;
    const long long* dst = edge_idx + E;

    // Workspace layout: sum_in[N*128] | sum_out[N*128] | cnt_in[N] | cnt_out[N]
    float* sum_in  = (float*)d_ws;
    float* sum_out = sum_in + (size_t)NN * DD;
    float* cnt_in  = sum_out + (size_t)NN * DD;
    float* cnt_out = cnt_in + NN;

    // 1) zero sums + counts
    int zn = NN * (2 * DD + 2);
    zero_ws_kernel<<<(zn + 255) / 256, 256, 0, stream>>>((float*)d_ws, zn);

    // 2) scatter-add
    long long at = (long long)E * 32;
    scatter_accum_kernel<<<(int)((at + 255) / 256), 256, 0, stream>>>(
        edge_attr, src, dst, sum_in, sum_out, cnt_in, cnt_out, E);

    // 3) normalize to means (both sum arrays and count arrays are contiguous)
    int nq = 2 * NN * 32;
    normalize_kernel<<<(nq + 255) / 256, 256, 0, stream>>>(sum_in, cnt_in, nq);

    // 4) fused WMMA GEMM over 32-edge tiles
    int tiles = (E + TILE_E - 1) / TILE_E;
    int blocks = tiles < 1024 ? tiles : 1024;
    fused_edge_gemm_kernel<<<blocks, 256, 0, stream>>>(
        edge_attr, src, dst, sum_in, sum_out,
        W_self_w, W_self_b, W_in_w, W_out_w, out, E, tiles);
}